// SynthesizerAttention_83141976917052
// MI455X (gfx1250) — compile-verified
//
#include <hip/hip_runtime.h>
#include <math.h>

// Problem constants (match reference)
#define B_   4
#define L_   2048
#define D_   1024
#define H_   16
#define HD_  64
#define K_   32
#define HHD  1024   // H_*HD_

typedef __attribute__((ext_vector_type(2))) float v2f;
typedef __attribute__((ext_vector_type(8))) float v8f;

// D = A(16x4, f32) * B(4x16, f32) + C(16x16, f32), full-f32 matrix pipe op.
__device__ __forceinline__ v8f wmma4(v2f a, v2f b, v8f c) {
    return __builtin_amdgcn_wmma_f32_16x16x4_f32(
        /*neg_a=*/false, a, /*neg_b=*/false, b,
        /*c_mod=*/(short)0, c, /*reuse_a=*/false, /*reuse_b=*/false);
}

// ---------------------------------------------------------------------------
// Tiled GEMM + per-column bias: C[M,N] = A[M,Kd] * Bm[Kd,N] + bias[N]
// Each wave computes a 32x32 C block as 2x2 WMMA subtiles: per k-step,
// 2 A-fragments + 2 B-fragments feed 4 WMMAs (each fragment reused twice),
// cutting the load:wmma ratio to 1.5:1.
// Fragment layouts (ISA 7.12.2, f32 16x16x4):
//   A 16x4 : lanes 0-15 -> M=lane, K=0,1 in v[0],v[1]; lanes 16-31 -> K=2,3
//   B 4x16 : VGPR0 = row K=0 (lanes 0-15) / K=2 (lanes 16-31); VGPR1 = K=1/K=3
//   D      : VGPR j -> M = j + 8*(lane>=16), N = lane%16
// ---------------------------------------------------------------------------
__global__ __launch_bounds__(256)
void gemm_bias_kernel(const float* __restrict__ A,
                      const float* __restrict__ Bm,
                      const float* __restrict__ bias,
                      float* __restrict__ C,
                      int M, int N, int Kd)
{
    const int lane = threadIdx.x & 31;
    const int wave = threadIdx.x >> 5;
    const int half = lane >> 4;
    const int l16  = lane & 15;

    const int nblk = N >> 5;                       // 32-wide N blocks
    const int blk  = blockIdx.x * 8 + wave;
    const int mt   = blk / nblk;
    const int nt   = blk % nblk;
    if (mt * 32 >= M) return;

    const int ka = half * 2;
    const float* Arow0 = A + (size_t)(mt * 32 + l16) * Kd + ka;
    const float* Arow1 = Arow0 + (size_t)16 * Kd;
    const float* Bcol  = Bm + (size_t)ka * N + nt * 32 + l16;

    v8f acc00 = {}, acc01 = {}, acc10 = {}, acc11 = {};
    for (int k0 = 0; k0 < Kd; k0 += 4) {
        v2f a0 = *(const v2f*)(Arow0 + k0);
        v2f a1 = *(const v2f*)(Arow1 + k0);
        v2f b0, b1;
        b0[0] = Bcol[(size_t)k0 * N];
        b0[1] = Bcol[(size_t)k0 * N + N];
        b1[0] = Bcol[(size_t)k0 * N + 16];
        b1[1] = Bcol[(size_t)k0 * N + N + 16];
        acc00 = wmma4(a0, b0, acc00);
        acc01 = wmma4(a0, b1, acc01);
        acc10 = wmma4(a1, b0, acc10);
        acc11 = wmma4(a1, b1, acc11);
    }

    const float bn0 = bias ? bias[nt * 32 + l16]      : 0.0f;
    const float bn1 = bias ? bias[nt * 32 + 16 + l16] : 0.0f;
    const int mb0 = mt * 32 + half * 8;
    const int mb1 = mb0 + 16;
    float* C00 = C + (size_t)mb0 * N + nt * 32 + l16;
    float* C10 = C + (size_t)mb1 * N + nt * 32 + l16;
#pragma unroll
    for (int j = 0; j < 8; ++j) {
        C00[(size_t)j * N]      = acc00[j] + bn0;
        C00[(size_t)j * N + 16] = acc01[j] + bn1;
        C10[(size_t)j * N]      = acc10[j] + bn0;
        C10[(size_t)j * N + 16] = acc11[j] + bn1;
    }
}

// ---------------------------------------------------------------------------
// Fused synthetic attention: one block per (head, 16-row query tile).
//   phase 1: scores[16, 2048] = r1[h, m0:m0+16, :32] @ r2[h, :32, :2048]  (WMMA)
//   phase 2: row-wise softmax entirely in LDS (320KB/WGP)
//   phase 3: y[b, m0:m0+16, h, :64] = attn @ v[b, :, h, :64]              (WMMA)
// The 16x2048 attention stripe never touches HBM.
// ---------------------------------------------------------------------------
#define LDSP 2052   // padded pitch: rows 8 apart land on different banks

__global__ __launch_bounds__(256)
void synth_attn_kernel(const float* __restrict__ r1,
                       const float* __restrict__ r2,
                       const float* __restrict__ v,
                       float* __restrict__ y)
{
    __shared__ float sc[16 * LDSP];     // ~128 KB score stripe
    __shared__ float red[16 * 16];
    __shared__ float rowmax[16];
    __shared__ float rowinv[16];

    const int lane = threadIdx.x & 31;
    const int wave = threadIdx.x >> 5;
    const int half = lane >> 4;
    const int l16  = lane & 15;

    const int h  = blockIdx.x >> 7;        // 128 row-tiles per head (L/16)
    const int mt = blockIdx.x & 127;
    const int m0 = mt * 16;

    // Preload all 8 A-fragments of r1[h, m0+m, 0..31] (K=32 -> 8 wmma k-steps)
    v2f afrag[8];
    {
        const float* a0 = r1 + ((size_t)h * L_ + m0 + l16) * K_ + half * 2;
#pragma unroll
        for (int s = 0; s < 8; ++s)
            afrag[s] = *(const v2f*)(a0 + 4 * s);
    }

    // ---- phase 1: synthetic scores -> LDS --------------------------------
    {
        const float* r2h = r2 + (size_t)h * K_ * L_;
        for (int ct = 0; ct < 16; ++ct) {              // each wave: 256 columns
            const int n0 = wave * 256 + ct * 16;
            const float* bp = r2h + (size_t)(half * 2) * L_ + n0 + l16;
            v8f acc = {};
#pragma unroll
            for (int s = 0; s < 8; ++s) {
                v2f b;
                b[0] = bp[(size_t)(4 * s) * L_];
                b[1] = bp[(size_t)(4 * s) * L_ + L_];
                acc = wmma4(afrag[s], b, acc);
            }
            const int rbase = half * 8;
#pragma unroll
            for (int j = 0; j < 8; ++j)
                sc[(rbase + j) * LDSP + n0 + l16] = acc[j];
        }
    }
    __syncthreads();

    // ---- phase 2: softmax over the 2048 key positions of each row --------
    {
        const int r  = threadIdx.x >> 4;   // 16 threads per row
        const int c0 = threadIdx.x & 15;

        float mx = -3.0e38f;
        for (int c = c0; c < L_; c += 16) mx = fmaxf(mx, sc[r * LDSP + c]);
        red[r * 16 + c0] = mx;
        __syncthreads();
        if (threadIdx.x < 16) {
            float mm = red[threadIdx.x * 16];
            for (int i = 1; i < 16; ++i) mm = fmaxf(mm, red[threadIdx.x * 16 + i]);
            rowmax[threadIdx.x] = mm;
        }
        __syncthreads();

        const float rm = rowmax[r];
        float s = 0.0f;
        for (int c = c0; c < L_; c += 16) {
            float e = __expf(sc[r * LDSP + c] - rm);
            sc[r * LDSP + c] = e;
            s += e;
        }
        red[r * 16 + c0] = s;
        __syncthreads();
        if (threadIdx.x < 16) {
            float ss = 0.0f;
            for (int i = 0; i < 16; ++i) ss += red[threadIdx.x * 16 + i];
            rowinv[threadIdx.x] = 1.0f / ss;
        }
        __syncthreads();

        const float ri = rowinv[r];
        for (int c = c0; c < L_; c += 16) sc[r * LDSP + c] *= ri;
    }
    __syncthreads();

    // ---- phase 3: y = attn @ v -------------------------------------------
    // Each wave owns 2 of the 16 (batch, HD-tile) combos.  The pair shares
    // the same batch and the same attention rows, so one ds_load_b64
    // A-fragment feeds two WMMAs per k-step.
    {
        const int base = wave * 2;
        const int b    = base >> 2;
        const int n0   = (base & 3) * 16;              // n1 = n0 + 16

        const float* vp = v + (((size_t)b * L_) * H_ + h) * HD_ + n0 + l16
                            + (size_t)(half * 2) * HHD;
        const float* ap = sc + l16 * LDSP + half * 2;  // attn A-frag from LDS

        v8f acc0 = {}, acc1 = {};
        for (int k0 = 0; k0 < L_; k0 += 4) {
            v2f a = *(const v2f*)(ap + k0);            // ds_load_b64, shared
            v2f bf0, bf1;
            bf0[0] = vp[(size_t)k0 * HHD];
            bf0[1] = vp[(size_t)k0 * HHD + HHD];
            bf1[0] = vp[(size_t)k0 * HHD + 16];
            bf1[1] = vp[(size_t)k0 * HHD + HHD + 16];
            acc0 = wmma4(a, bf0, acc0);
            acc1 = wmma4(a, bf1, acc1);
        }

        const int mbase = m0 + half * 8;
        float* yp = y + (((size_t)b * L_ + mbase) * H_ + h) * HD_ + n0 + l16;
#pragma unroll
        for (int j = 0; j < 8; ++j) {
            yp[(size_t)j * HHD]      = acc0[j];
            yp[(size_t)j * HHD + 16] = acc1[j];
        }
    }
}

// ---------------------------------------------------------------------------
extern "C" void kernel_launch(void* const* d_in, const int* in_sizes, int n_in,
                              void* d_out, int out_size, void* d_ws, size_t ws_size,
                              hipStream_t stream)
{
    // setup_inputs order: inputs_q, inputs_kv, Wv, bv, r1, r2, Wo, bo
    const float* inputs_kv = (const float*)d_in[1];  // inputs_q unused (ignore_dot_product)
    const float* Wv = (const float*)d_in[2];
    const float* bv = (const float*)d_in[3];
    const float* r1 = (const float*)d_in[4];
    const float* r2 = (const float*)d_in[5];
    const float* Wo = (const float*)d_in[6];
    const float* bo = (const float*)d_in[7];
    float* out = (float*)d_out;

    float* v_ws = (float*)d_ws;                                // [B,L,H,HD] 32 MB
    float* y_ws = v_ws + (size_t)B_ * L_ * H_ * HD_;           // [B,L,H,HD] 32 MB

    const dim3 blk(256);
    const int M = B_ * L_;                                     // 8192
    const int gemm_blocks = (M / 32) * (D_ / 32) / 8;          // 8 wave-blocks each

    // 1) value projection: v = inputs_kv @ Wv + bv
    gemm_bias_kernel<<<gemm_blocks, blk, 0, stream>>>(inputs_kv, Wv, bv, v_ws,
                                                      M, HHD, D_);
    // 2) fused synthetic scores + softmax + weighted value sum
    synth_attn_kernel<<<H_ * (L_ / 16), blk, 0, stream>>>(r1, r2, v_ws, y_ws);

    // 3) output projection: out = y @ Wo + bo
    gemm_bias_kernel<<<gemm_blocks, blk, 0, stream>>>(y_ws, Wo, bo, out,
                                                      M, D_, HHD);
}